// SquaredReLUAttentionLayer_11020886082256
// MI455X (gfx1250) — compile-verified
//
#include <hip/hip_runtime.h>

// Squared-ReLU attention, flash-style streaming, f32 WMMA (CDNA5 / gfx1250).
// out = relu(scale * Q K^T - |i-j|)^2 @ V
// Double-buffered async global->LDS staging of K/V tiles (ASYNCcnt path).

typedef __attribute__((ext_vector_type(2))) float v2f;
typedef __attribute__((ext_vector_type(8))) float v8f;
typedef int v4i __attribute__((vector_size(16)));   // builtin's b128 payload type

#define ZDIM 2
#define HDIMS 8
#define N_CTX 2048
#define DHEAD 64
#define BLK_J 16
#define NBLK (N_CTX / BLK_J)
#define WAVES_PER_WG 8
#define BLOCK_M (WAVES_PER_WG * 16)   // 128 query rows per workgroup

// LDS row paddings chosen for bank-conflict-free fragment gathers (64 banks):
#define KPAD 68   // K tile: B-frag b64 gather hits all 64 banks exactly once
#define VPAD 72   // V tile: b32 gathers, low/high half-wave bank sets disjoint
#define PPAD 18   // P transpose scratch: rows 8-byte aligned, reads conflict-free

#if __has_builtin(__builtin_amdgcn_global_load_async_to_lds_b128)
#define USE_ASYNC_LDS 1
#else
#define USE_ASYNC_LDS 0
#endif

#if __has_builtin(__builtin_amdgcn_s_wait_asynccnt)
#define WAIT_ASYNC0() __builtin_amdgcn_s_wait_asynccnt(0)
#else
#define WAIT_ASYNC0() asm volatile("s_wait_asynccnt 0x0" ::: "memory")
#endif

// Param types per hipcc diagnostic: (v4i AS1*, v4i AS3*, imm int, imm int)
#define AS_GLOBAL(p) ((__attribute__((address_space(1))) v4i*)(p))
#define AS_LDS(p)    ((__attribute__((address_space(3))) v4i*)(p))

__global__ __launch_bounds__(256, 1)
void sqrelu_attn_f32_wmma(const float* __restrict__ Q,
                          const float* __restrict__ K,
                          const float* __restrict__ V,
                          const float* __restrict__ scale_p,
                          float* __restrict__ O)
{
    __shared__ float lds_k[2][BLK_J][KPAD];   // double-buffered K tile
    __shared__ float lds_v[2][BLK_J][VPAD];   // double-buffered V tile
    __shared__ float lds_p[WAVES_PER_WG][16][PPAD];

    const int tid    = threadIdx.x;
    const int wave   = tid >> 5;
    const int lane   = tid & 31;
    const int laneLo = lane & 15;   // N / M index within fragment
    const int laneHi = lane >> 4;   // selects K-subgroup / M+8 half

    const int    bh    = blockIdx.y;                 // z*H + h
    const size_t base  = (size_t)bh * N_CTX * DHEAD; // head offset (elements)
    const int    qRow0 = blockIdx.x * BLOCK_M + wave * 16;

    const float scale = *scale_p;

    // Per-thread slice of the cooperative 16x64 tile stage (one b128 each).
    const int sr = (tid * 4) >> 6;   // row in tile
    const int sc = (tid * 4) & 63;   // col (multiple of 4 -> 16B aligned)
    const float* kp0 = K + base + (size_t)sr * DHEAD + sc;
    const float* vp0 = V + base + (size_t)sr * DHEAD + sc;

    // Stage K/V block starting at key j0 into LDS buffer `buf`.
    auto stage = [&](int buf, int j0) {
#if USE_ASYNC_LDS
        __builtin_amdgcn_global_load_async_to_lds_b128(
            AS_GLOBAL(kp0 + (size_t)j0 * DHEAD), AS_LDS(&lds_k[buf][sr][sc]), 0, 0);
        __builtin_amdgcn_global_load_async_to_lds_b128(
            AS_GLOBAL(vp0 + (size_t)j0 * DHEAD), AS_LDS(&lds_v[buf][sr][sc]), 0, 0);
#else
        *(float4*)&lds_k[buf][sr][sc] = *(const float4*)(kp0 + (size_t)j0 * DHEAD);
        *(float4*)&lds_v[buf][sr][sc] = *(const float4*)(vp0 + (size_t)j0 * DHEAD);
#endif
        // deeper prefetch: pull block j0+BLK_J toward L2 while j0 is in flight
        if (j0 + BLK_J < N_CTX) {
            __builtin_prefetch(kp0 + (size_t)(j0 + BLK_J) * DHEAD, 0, 1);
            __builtin_prefetch(vp0 + (size_t)(j0 + BLK_J) * DHEAD, 0, 1);
        }
    };

    // ---- Load Q tile (16 x 64) once, directly in f32 A-fragment layout ----
    // A 16x4 f32: lane L (L<16): M=L, regs = K {0,1}; lane L>=16: M=L-16, K {2,3}
    v2f qf[16];
    {
        const float* qrow = Q + base + (size_t)(qRow0 + laneLo) * DHEAD;
        #pragma unroll
        for (int c = 0; c < 16; ++c) {
            const int kk = 4 * c + 2 * laneHi;
            qf[c] = *(const v2f*)(qrow + kk);   // 8B-aligned (kk even)
        }
    }

    // ---- Output accumulators: 16(M) x 64(D) as 4 WMMA C/D tiles ----
    v8f acc[4] = {};

    stage(0, 0);   // prologue: fill buffer 0

    for (int jb = 0; jb < NBLK; ++jb) {
        const int j0  = jb * BLK_J;
        const int cur = jb & 1;

#if USE_ASYNC_LDS
        WAIT_ASYNC0();        // my async writes into buf `cur` have landed
#endif
        __syncthreads();      // everyone's writes visible; everyone done
                              // reading buf cur^1 (last used in jb-1)

        if (jb + 1 < NBLK) stage(cur ^ 1, j0 + BLK_J);   // overlap with compute

        // ---- S = Q * K^T  (16x16), K-dim = 64 -> 16 x WMMA f32 16x16x4 ----
        v8f s = {};
        #pragma unroll
        for (int c = 0; c < 16; ++c) {
            const int kk = 4 * c + 2 * laneHi;
            // B[k][n] = K[j0+n][k]; n = laneLo across lanes, k in regs
            v2f b = *(const v2f*)&lds_k[cur][laneLo][kk];   // 8B-aligned
            s = __builtin_amdgcn_wmma_f32_16x16x4_f32(
                    false, qf[c], false, b, (short)0, s, false, false);
        }

        // ---- Epilogue: p = relu(scale*s - |i-j|)^2, write to transpose LDS -
        // C/D layout: reg e, lanes 0-15 -> (M=e, N=lane); lanes 16-31 -> M=e+8
        #pragma unroll
        for (int e = 0; e < 8; ++e) {
            const int i = qRow0 + e + 8 * laneHi;
            const int j = j0 + laneLo;
            float sv = s[e] * scale - fabsf((float)(i - j));
            sv = fmaxf(sv, 0.0f);
            lds_p[wave][e + 8 * laneHi][laneLo] = sv * sv;
        }
        // same-wave LDS RAW: compiler inserts s_wait_dscnt; no barrier needed

        // ---- O += P * V : K-dim = 16 -> 4 chunks x 4 D-tiles of WMMA ----
        #pragma unroll
        for (int c = 0; c < 4; ++c) {
            const int kk = 4 * c + 2 * laneHi;
            // A[m][k] = P[m][k]: m = laneLo, k in regs (A layout)
            v2f a = *(const v2f*)&lds_p[wave][laneLo][kk];  // 8B-aligned (PPAD even)
            #pragma unroll
            for (int t = 0; t < 4; ++t) {
                // B[k][n] = V[j0+k][t*16+n]
                v2f b;
                b.x = lds_v[cur][kk]    [t * 16 + laneLo];
                b.y = lds_v[cur][kk + 1][t * 16 + laneLo];
                acc[t] = __builtin_amdgcn_wmma_f32_16x16x4_f32(
                             false, a, false, b, (short)0, acc[t], false, false);
            }
        }
    }

    // ---- Store 16x64 output tile (write-once: nontemporal) ----
    #pragma unroll
    for (int t = 0; t < 4; ++t) {
        #pragma unroll
        for (int e = 0; e < 8; ++e) {
            const int i = qRow0 + e + 8 * laneHi;
            const int d = t * 16 + laneLo;
            __builtin_nontemporal_store(acc[t][e], O + base + (size_t)i * DHEAD + d);
        }
    }
}

extern "C" void kernel_launch(void* const* d_in, const int* in_sizes, int n_in,
                              void* d_out, int out_size, void* d_ws, size_t ws_size,
                              hipStream_t stream) {
    (void)in_sizes; (void)n_in; (void)out_size; (void)d_ws; (void)ws_size;
    const float* q     = (const float*)d_in[0];
    const float* k     = (const float*)d_in[1];
    const float* v     = (const float*)d_in[2];
    const float* scale = (const float*)d_in[3];
    float* out = (float*)d_out;

    dim3 grid(N_CTX / BLOCK_M, ZDIM * HDIMS);  // (16, 16)
    dim3 block(256);                           // 8 wave32 waves
    sqrelu_attn_f32_wmma<<<grid, block, 0, stream>>>(q, k, v, scale, out);
}